// resize_3D_21500606284455
// MI455X (gfx1250) — compile-verified
//
#include <hip/hip_runtime.h>

// ---------------------------------------------------------------------------
// Trilinear 2x upsample (TF1 legacy align: src = dst*0.5, frac in {0, 0.5})
// x: (4, 32, 32, 32, 32) f32  ->  out: (4, 64, 64, 64, 32) f32
//
// Per block (b, d0, h0):
//   1) async global->LDS stage of the 4 corner rows (d,h) of 32w x 32c
//   2) branch-free combine into 4 tables T[fd*2+fh] (weights 1/0.5/0.25)
//   3) per wave: O(64x16) = A(64x32) @ T(32x16) via V_WMMA_F32_16X16X4_F32,
//      using A's banded sparsity: tile i only needs K-chunks {2i, 2i+1, 2i+2}
//      -> 11 WMMAs per wave, all A fragments synthesized branchlessly.
// ---------------------------------------------------------------------------

typedef float v2f __attribute__((ext_vector_type(2)));
typedef float v4f __attribute__((ext_vector_type(4)));
typedef float v8f __attribute__((ext_vector_type(8)));
typedef int   v4i __attribute__((ext_vector_type(4)));

#if __has_builtin(__builtin_amdgcn_wmma_f32_16x16x4_f32)
#define USE_WMMA 1
#else
#define USE_WMMA 0
#endif

#if __has_builtin(__builtin_amdgcn_global_load_async_to_lds_b128)
#define USE_ASYNC 1
#else
#define USE_ASYNC 0
#endif

// Branchless A[ow][k] fragment for one tile row block i.
// Per-lane precomputed: m2 = (lane&15)>>1, wlo = odd?0.5:1.0, whi = odd?0.5:0.
// lo = 8i + m2, hi = min(lo+1, 31); entry = (k==lo)*wlo + (k==hi)*whi.
// (even ow -> 1 at lo; odd ow -> 0.5 at lo and hi; ow==63 -> lo==hi==31 -> 1.)
__device__ __forceinline__ v2f afrag(int i, int k0, int m2, float wlo, float whi) {
  const int lo = 8 * i + m2;
  const int hi = (lo < 31) ? (lo + 1) : 31;
  v2f af;
  af.x = ((k0     == lo) ? wlo : 0.0f) + ((k0     == hi) ? whi : 0.0f);
  af.y = ((k0 + 1 == lo) ? wlo : 0.0f) + ((k0 + 1 == hi) ? whi : 0.0f);
  return af;
}

__global__ __launch_bounds__(256) void upsample3d_wmma_kernel(
    const float* __restrict__ x, float* __restrict__ out) {
  __shared__ float ldsRaw[4 * 1024];   // 4 corner rows, 32w x 32c each (16 KB)
  __shared__ float ldsT[4 * 1024];     // 4 combined tables, one per (fd,fh)

  const int t   = threadIdx.x;
  const int bid = blockIdx.x;           // 4096 blocks = 4 * 32 * 32
  const int b   = bid >> 10;
  const int d0  = (bid >> 5) & 31;
  const int h0  = bid & 31;
  const int d1  = (d0 < 31) ? d0 + 1 : 31;
  const int h1  = (h0 < 31) ? h0 + 1 : 31;

  const float* base = x + (size_t)b * 1048576u;   // 32*32*32*32
  const int dsel[4] = {d0, d0, d1, d1};
  const int hsel[4] = {h0, h1, h0, h1};

  // ---- Phase 0: stage the 4 corner rows into LDS --------------------------
#if USE_ASYNC
  #pragma unroll
  for (int r = 0; r < 4; ++r) {
    const float* g = base + dsel[r] * 32768 + hsel[r] * 1024 + t * 4;
    float* l = &ldsRaw[r * 1024 + t * 4];
    __builtin_amdgcn_global_load_async_to_lds_b128((v4i*)g, (v4i*)l, 0, 0);
  }
  #if __has_builtin(__builtin_amdgcn_s_wait_asynccnt)
  __builtin_amdgcn_s_wait_asynccnt(0);
  #else
  asm volatile("s_wait_asynccnt 0" ::: "memory");
  #endif
#else
  #pragma unroll
  for (int r = 0; r < 4; ++r) {
    const v4f* g = reinterpret_cast<const v4f*>(base + dsel[r] * 32768 + hsel[r] * 1024);
    reinterpret_cast<v4f*>(ldsRaw)[r * 256 + t] = g[t];
  }
#endif
  __syncthreads();

  // ---- Phase 1: branch-free combine into the 4 (fd,fh) tables -------------
  // T0 = L00 ; T1 = 0.5*(L00+L01) ; T2 = 0.5*(L00+L10) ; T3 = 0.25*sum
  {
    const v4f* r4 = reinterpret_cast<const v4f*>(ldsRaw);
    v4f* t4 = reinterpret_cast<v4f*>(ldsT);
    const v4f l0 = r4[t];
    const v4f l1 = r4[256 + t];
    const v4f l2 = r4[512 + t];
    const v4f l3 = r4[768 + t];
    t4[t]       = l0;
    t4[256 + t] = 0.5f * (l0 + l1);
    t4[512 + t] = 0.5f * (l0 + l2);
    t4[768 + t] = 0.25f * ((l0 + l1) + (l2 + l3));
  }
  __syncthreads();

  // ---- Phase 2: W-axis resize as WMMA matmul per (od, oh) slab ------------
  const int lane = t & 31;
  const int wid  = t >> 5;              // 8 waves
  const int fd   = wid & 1;             // od = 2*d0 + fd
  const int fh   = (wid >> 1) & 1;      // oh = 2*h0 + fh
  const int j    = wid >> 2;            // c half: cols j*16 .. j*16+15

  const int od  = 2 * d0 + fd;
  const int oh  = 2 * h0 + fh;
  const int col = j * 16 + (lane & 15);
  const int khalf = (lane >> 4) * 2;    // A/B frag K layout: lanes16-31 hold K+2

  const float* T = ldsT + (2 * fd + fh) * 1024;   // wave-uniform table
  const size_t obase = (size_t)b * 8388608u + (size_t)od * 131072u + (size_t)oh * 2048u;

#if USE_WMMA
  const int   m2  = (lane & 15) >> 1;
  const int   odd = lane & 1;           // parity of the output row this lane owns
  const float wlo = odd ? 0.5f : 1.0f;
  const float whi = odd ? 0.5f : 0.0f;

  v8f z = {0.f, 0.f, 0.f, 0.f, 0.f, 0.f, 0.f, 0.f};
  v8f acc[4] = {z, z, z, z};

  #pragma unroll
  for (int kc = 0; kc < 8; ++kc) {      // K = 32 in chunks of 4
    const int k0 = kc * 4 + khalf;
    v2f bf;                             // B frag (4x16): VGPR0=K{0,2}, VGPR1=K{1,3}
    bf.x = T[k0 * 32 + col];
    bf.y = T[(k0 + 1) * 32 + col];
    {   // primary tile for this K-chunk
      const int i = kc >> 1;
      acc[i] = __builtin_amdgcn_wmma_f32_16x16x4_f32(
          false, afrag(i, k0, m2, wlo, whi), false, bf, (short)0, acc[i], false, false);
    }
    if ((kc & 1) == 0 && kc >= 2) {     // trailing tile: its w == 8i+8 edge
      const int i = (kc >> 1) - 1;
      acc[i] = __builtin_amdgcn_wmma_f32_16x16x4_f32(
          false, afrag(i, k0, m2, wlo, whi), false, bf, (short)0, acc[i], false, false);
    }
  }

  // C/D layout: VGPR r -> rows r (lanes 0-15) and r+8 (lanes 16-31)
  #pragma unroll
  for (int i = 0; i < 4; ++i) {
    #pragma unroll
    for (int r = 0; r < 8; ++r) {
      const int ow = i * 16 + r + ((lane >> 4) << 3);
      __builtin_nontemporal_store(acc[i][r], &out[obase + (size_t)ow * 32 + col]);
    }
  }
#else
  // VALU fallback: direct lerp along W from the combined table
  #pragma unroll
  for (int i = 0; i < 4; ++i) {
    #pragma unroll
    for (int r = 0; r < 8; ++r) {
      const int ow = i * 16 + r + ((lane >> 4) << 3);
      const int lo = ow >> 1;
      int hi = lo + 1; if (hi > 31) hi = 31;
      float v = T[lo * 32 + col];
      if (ow & 1) v = 0.5f * (v + T[hi * 32 + col]);
      __builtin_nontemporal_store(v, &out[obase + (size_t)ow * 32 + col]);
    }
  }
#endif
}

extern "C" void kernel_launch(void* const* d_in, const int* in_sizes, int n_in,
                              void* d_out, int out_size, void* d_ws, size_t ws_size,
                              hipStream_t stream) {
  const float* x = (const float*)d_in[0];
  float* out = (float*)d_out;
  // 4 batches * 32 d-cells * 32 h-cells = 4096 blocks; 256 threads (8 waves)
  upsample3d_wmma_kernel<<<dim3(4096), dim3(256), 0, stream>>>(x, out);
}